// TopKMoEGate_10917806866936
// MI455X (gfx1250) — compile-verified
//
#include <hip/hip_runtime.h>
#include <hip/hip_bf16.h>

typedef __attribute__((ext_vector_type(2))) float v2f;
typedef __attribute__((ext_vector_type(8))) float v8f;

namespace {
constexpr int kD   = 2048;   // hidden dim (K of GEMM)
constexpr int kE   = 64;     // experts    (N of GEMM)
constexpr int kTok = 8192;   // B*S tokens (M of GEMM)
constexpr int kTop = 2;

constexpr int kWaves        = 4;           // waves per block (wave32)
constexpr int kMT           = 2;           // 16-row M tiles per wave
constexpr int kRowsPerWave  = 16 * kMT;    // 32
constexpr int kRowsPerBlock = kWaves * kRowsPerWave;  // 128
}

// ---------------------------------------------------------------------------
// Kernel 1: logits_noisy[t][e] = sum_k x[t][k] * W[e][k] + noise[t][e]*nw[e]
// Wave32 WMMA fp32: each wave owns 32 rows x 64 experts (2 M-tiles x 4 N-tiles).
// A 16x4 fp32 layout: lane l(0-15) row=l, K=kb+{0,1}; lane l+16 row=l, K=kb+{2,3}
//   -> per-lane contiguous float2 load at x + row*kD + kb + 2*(lane>>4).
// B 4x16 fp32 layout mirrors with N=lane&15 -> contiguous float2 from W rows.
// ---------------------------------------------------------------------------
__global__ __launch_bounds__(kWaves * 32, 1)
void moe_gate_logits(const float* __restrict__ x,
                     const float* __restrict__ W,
                     const float* __restrict__ nw,
                     const float* __restrict__ noise,
                     float* __restrict__ logits)
{
    const int lane = threadIdx.x & 31;
    const int wave = threadIdx.x >> 5;
    const int g    = lane >> 4;      // half-wave group: selects K sub-pair
    const int l    = lane & 15;      // row (A) / expert (B) within tile
    const int rowBase = blockIdx.x * kRowsPerBlock + wave * kRowsPerWave;

    const float* a0 = x + (size_t)(rowBase + l) * kD + 2 * g;
    const float* a1 = a0 + (size_t)16 * kD;
    const float* b0 = W + (size_t)l * kD + 2 * g;
    const float* b1 = b0 + (size_t)16 * kD;
    const float* b2 = b0 + (size_t)32 * kD;
    const float* b3 = b0 + (size_t)48 * kD;

    v8f acc[kMT][4];
#pragma unroll
    for (int m = 0; m < kMT; ++m)
#pragma unroll
        for (int n = 0; n < 4; ++n)
            acc[m][n] = (v8f){0.f, 0.f, 0.f, 0.f, 0.f, 0.f, 0.f, 0.f};

#pragma unroll 4
    for (int kb = 0; kb < kD; kb += 4) {
        const v2f av0 = *(const v2f*)(a0 + kb);
        const v2f av1 = *(const v2f*)(a1 + kb);
        const v2f bv0 = *(const v2f*)(b0 + kb);
        const v2f bv1 = *(const v2f*)(b1 + kb);
        const v2f bv2 = *(const v2f*)(b2 + kb);
        const v2f bv3 = *(const v2f*)(b3 + kb);

        acc[0][0] = __builtin_amdgcn_wmma_f32_16x16x4_f32(false, av0, false, bv0,
                                                          (short)0, acc[0][0], false, false);
        acc[0][1] = __builtin_amdgcn_wmma_f32_16x16x4_f32(false, av0, false, bv1,
                                                          (short)0, acc[0][1], false, false);
        acc[0][2] = __builtin_amdgcn_wmma_f32_16x16x4_f32(false, av0, false, bv2,
                                                          (short)0, acc[0][2], false, false);
        acc[0][3] = __builtin_amdgcn_wmma_f32_16x16x4_f32(false, av0, false, bv3,
                                                          (short)0, acc[0][3], false, false);
        acc[1][0] = __builtin_amdgcn_wmma_f32_16x16x4_f32(false, av1, false, bv0,
                                                          (short)0, acc[1][0], false, false);
        acc[1][1] = __builtin_amdgcn_wmma_f32_16x16x4_f32(false, av1, false, bv1,
                                                          (short)0, acc[1][1], false, false);
        acc[1][2] = __builtin_amdgcn_wmma_f32_16x16x4_f32(false, av1, false, bv2,
                                                          (short)0, acc[1][2], false, false);
        acc[1][3] = __builtin_amdgcn_wmma_f32_16x16x4_f32(false, av1, false, bv3,
                                                          (short)0, acc[1][3], false, false);
    }

    // Epilogue: fuse noise injection, spill logits to workspace.
    // C/D layout: VGPR r, lanes 0-15 -> (M=r, N=lane); lanes 16-31 -> (M=r+8, N=lane-16)
#pragma unroll
    for (int n = 0; n < 4; ++n) {
        const int e   = n * 16 + l;
        const float we = nw[e];
#pragma unroll
        for (int m = 0; m < kMT; ++m) {
#pragma unroll
            for (int r = 0; r < 8; ++r) {
                const int row = rowBase + m * 16 + r + 8 * g;
                const size_t off = (size_t)row * kE + e;
                logits[off] = acc[m][n][r] + noise[off] * we;
            }
        }
    }
}

// ---------------------------------------------------------------------------
// Kernel 2: per-token top-2 over 64 experts, softmax of the 2 winners,
// scatter into zeroed gated row; emit top_idx (as float) and top_vals.
// ---------------------------------------------------------------------------
__global__ __launch_bounds__(256)
void moe_gate_topk(const float* __restrict__ logits,
                   float* __restrict__ gated,
                   float* __restrict__ top_idx,
                   float* __restrict__ top_vals)
{
    const int t = blockIdx.x * blockDim.x + threadIdx.x;
    if (t >= kTok) return;

    const float* row = logits + (size_t)t * kE;
    float v1 = -INFINITY, v2 = -INFINITY;
    int i1 = 0, i2 = 0;
#pragma unroll 8
    for (int e = 0; e < kE; ++e) {
        const float v = row[e];
        if (v > v1)      { v2 = v1; i2 = i1; v1 = v; i1 = e; }
        else if (v > v2) { v2 = v;  i2 = e; }
    }

    const float e2 = __expf(v2 - v1);          // v1 >= v2, numerically safe
    const float p1 = 1.0f / (1.0f + e2);
    const float p2 = 1.0f - p1;

    float* grow = gated + (size_t)t * kE;
#pragma unroll
    for (int e = 0; e < kE; ++e) grow[e] = 0.0f;
    grow[i1] = p1;
    grow[i2] = p2;

    top_idx[(size_t)t * kTop + 0]  = (float)i1;
    top_idx[(size_t)t * kTop + 1]  = (float)i2;
    top_vals[(size_t)t * kTop + 0] = v1;
    top_vals[(size_t)t * kTop + 1] = v2;
}

// ---------------------------------------------------------------------------
// Launch: inputs are (x, W, noise_weight, noise, k). Output layout is
// [gated (B*S*E) | top_idx (B*S*K) | top_vals (B*S*K)] flattened floats.
// d_ws holds the noisy logits (8192*64*4 = 2 MB).
// ---------------------------------------------------------------------------
extern "C" void kernel_launch(void* const* d_in, const int* in_sizes, int n_in,
                              void* d_out, int out_size, void* d_ws, size_t ws_size,
                              hipStream_t stream)
{
    const float* x     = (const float*)d_in[0];
    const float* W     = (const float*)d_in[1];
    const float* nw    = (const float*)d_in[2];
    const float* noise = (const float*)d_in[3];
    // d_in[4] is k (== 2), compile-time constant here.

    float* out      = (float*)d_out;
    float* gated    = out;
    float* top_idx  = out + (size_t)kTok * kE;
    float* top_vals = top_idx + (size_t)kTok * kTop;
    float* logits   = (float*)d_ws;

    dim3 grid1(kTok / kRowsPerBlock);   // 64 blocks
    dim3 block1(kWaves * 32);           // 128 threads = 4 wave32
    moe_gate_logits<<<grid1, block1, 0, stream>>>(x, W, nw, noise, logits);

    dim3 grid2((kTok + 255) / 256);     // 32 blocks
    dim3 block2(256);
    moe_gate_topk<<<grid2, block2, 0, stream>>>(logits, gated, top_idx, top_vals);
}